// ANPM_5583457485031
// MI455X (gfx1250) — compile-verified
//
#include <hip/hip_runtime.h>
#include <math.h>

// MI455X (gfx1250, wave32). Reference reduces to 3 memory-bound streaming
// passes over x1/x2 (819MB): ~2.46GB total -> ~105us at 23.3 TB/s peak.
// Row dot-products run on V_WMMA_F32_16X16X4_F32 (cross-lane reduce in the
// matrix unit); x tiles staged in LDS (132-float pitch => conflict-free
// ds_load_b64 B-operand feed), staged with GLOBAL_LOAD_ASYNC_TO_LDS_B128
// when the toolchain exposes it (ASYNCcnt path), coalesced b128 otherwise.

typedef float v2f __attribute__((ext_vector_type(2)));
typedef float v8f __attribute__((ext_vector_type(8)));

// Signature of __builtin_amdgcn_global_load_async_to_lds_b128 (from compiler
// diagnostic): (int __vector(4) AS(1)*, int __vector(4) AS(3)*, imm, imm)
typedef int v4i_ls __attribute__((vector_size(16)));
typedef v4i_ls __attribute__((address_space(1)))* anpm_gp128_t;
typedef v4i_ls __attribute__((address_space(3)))* anpm_lp128_t;

#define NPAIR 16   // (tensor 0/1) x (batch 0..7)

#if defined(__HIP_DEVICE_COMPILE__) && __has_builtin(__builtin_amdgcn_global_load_async_to_lds_b128)
#define HAVE_ASYNC_LDS 1
#else
#define HAVE_ASYNC_LDS 0
#endif

static __device__ __forceinline__ void anpm_wait_async() {
#if __has_builtin(__builtin_amdgcn_s_wait_asynccnt)
  __builtin_amdgcn_s_wait_asynccnt(0);
#else
  asm volatile("s_wait_asynccnt 0" ::: "memory");
#endif
}

// ---------------------------------------------------------------------------
// Pass 1: per-(t,b) column sum of x (for mean). 256 thr = 8 waves; each wave
// streams whole rows (32 lanes x float4 = 512B coalesced per instruction).
// ---------------------------------------------------------------------------
__global__ __launch_bounds__(256) void anpm_mean_kernel(
    const float* __restrict__ x1, const float* __restrict__ x2,
    int N, int nwg, float* __restrict__ meanPart)
{
  int blk = blockIdx.x, pair = blk / nwg, wg = blk % nwg;
  const float* x = ((pair >> 3) ? x2 : x1) + (size_t)(pair & 7) * N * 128;
  int tid = threadIdx.x, lane = tid & 31, warp = tid >> 5;

  float4 acc = make_float4(0.f, 0.f, 0.f, 0.f);
  int stride = nwg * 8;
  for (int row = wg * 8 + warp; row < N; row += stride) {
    float4 v = *reinterpret_cast<const float4*>(x + (size_t)row * 128 + lane * 4);
    acc.x += v.x; acc.y += v.y; acc.z += v.z; acc.w += v.w;
  }
  __shared__ __align__(16) float sm[8 * 128];
  *reinterpret_cast<float4*>(&sm[warp * 128 + lane * 4]) = acc;
  __syncthreads();
  if (tid < 128) {
    float s = 0.f;
    for (int w = 0; w < 8; ++w) s += sm[w * 128 + tid];
    meanPart[(size_t)pair * nwg * 128 + (size_t)wg * 128 + tid] = s;
  }
}

// ---------------------------------------------------------------------------
// Stage: reduce partial sums -> mean vector(s); h = tanh(m @ W_att[i]);
// w = Va + Wt_i @ h ; c = Vb.h + b_att[i]. Writes wAll/cAll slots.
// vecs==1: both heads use the same mean; vecs==2: head i uses vector i.
// ---------------------------------------------------------------------------
__global__ __launch_bounds__(128) void anpm_stage_kernel(
    const float* __restrict__ part, int nwg, int vecs, float invN,
    const float* __restrict__ W_att, const float* __restrict__ V_att,
    const float* __restrict__ Wt_att, const float* __restrict__ b_att,
    float* __restrict__ wAll, float* __restrict__ cAll, int slot0)
{
  int pair = blockIdx.x, tid = threadIdx.x;
  __shared__ float mv[2][128];
  __shared__ float hh[128];
  __shared__ float red[128];

  int stride = vecs * 128;
  for (int v = 0; v < vecs; ++v) {
    float s = 0.f;
    const float* p = part + (size_t)pair * nwg * stride + v * 128 + tid;
    for (int w = 0; w < nwg; ++w) s += p[(size_t)w * stride];
    mv[v][tid] = s * invN;
  }
  __syncthreads();

  for (int i = 0; i < 2; ++i) {
    const float* m = mv[(vecs == 2) ? i : 0];
    const float* Wi = W_att + (size_t)i * 16384;
    float a = 0.f;
    for (int d = 0; d < 128; ++d) a = fmaf(m[d], Wi[d * 128 + tid], a);  // coalesced over tid
    hh[tid] = tanhf(a);
    __syncthreads();

    const float* Ti = Wt_att + (size_t)i * 16384;   // Wt_att[i,0,:,:]
    float u = 0.f;
    for (int e = 0; e < 128; ++e) u = fmaf(Ti[(size_t)tid * 128 + e], hh[e], u);
    wAll[((size_t)pair * 4 + slot0 + i) * 128 + tid] = V_att[i * 256 + tid] + u;

    red[tid] = V_att[i * 256 + 128 + tid] * hh[tid];
    __syncthreads();
    for (int off = 64; off > 0; off >>= 1) {
      if (tid < off) red[tid] += red[tid + off];
      __syncthreads();
    }
    if (tid == 0) cAll[pair * 4 + slot0 + i] = red[0] + b_att[i];
    __syncthreads();
  }
}

// ---------------------------------------------------------------------------
// Tile staging helpers (per-wave 16x128 tile, pitch 132 floats).
// ---------------------------------------------------------------------------
static __device__ __forceinline__ void anpm_stage_tile_masked(
    const float* __restrict__ x, float* tileLds, int base, int N, int lane)
{
#pragma unroll
  for (int r = 0; r < 16; ++r) {
    int row = base + r;
    float4 v = make_float4(0.f, 0.f, 0.f, 0.f);
    if (row < N) v = *reinterpret_cast<const float4*>(x + (size_t)row * 128 + lane * 4);
    *reinterpret_cast<float4*>(tileLds + r * 132 + lane * 4) = v;
  }
  __builtin_amdgcn_wave_barrier();
}

// ---------------------------------------------------------------------------
// WMMA streaming pass. One wave owns a 16x128 x-tile in LDS. 32 steps of
// V_WMMA_F32_16X16X4_F32 accumulate C[j][row] = x_row . w_j ; all dots for
// row m land in lane m, so att/coef is computed in-lane and broadcast with
// v_readlane, then the weighted row-sum is accumulated with float4 FMAs from
// LDS. Deterministic two-stage partial reduction (no float atomics).
// ---------------------------------------------------------------------------
template <int PASS>
__global__ __launch_bounds__(128) void anpm_pass_kernel(
    const float* __restrict__ x1, const float* __restrict__ x2,
    int N, int nwg,
    const float* __restrict__ wAll, const float* __restrict__ cAll,
    const float* __restrict__ U_att, float* __restrict__ part)
{
  constexpr int NW = (PASS == 2) ? 2 : 4;
  int blk = blockIdx.x, pair = blk / nwg, wg = blk % nwg;
  const float* x = ((pair >> 3) ? x2 : x1) + (size_t)(pair & 7) * N * 128;
  int tid = threadIdx.x, lane = tid & 31, warp = tid >> 5;

  __shared__ __align__(16) float sm[4 * 2112];   // 4 waves x (16 rows x 132 pitch)
  float* tileLds = sm + warp * 2112;

  const float* wv = wAll + (size_t)pair * 4 * 128;
  float c0h = cAll[pair * 4 + 0], c1h = cAll[pair * 4 + 1];
  float c2h = cAll[pair * 4 + 2], c3h = cAll[pair * 4 + 3];  // unused when PASS==2
  float U0 = U_att[0], U1 = U_att[1];

  int wavesPerPair = nwg * 4;
  int waveId = wg * 4 + warp;
  int numTiles = (N + 15) >> 4;

  int j = lane & 15;
  int half = (lane >= 16) ? 2 : 0;
  float amsk = (j < NW) ? 1.0f : 0.0f;
  int jj = (j < NW) ? j : 0;

  float4 acc0 = make_float4(0.f, 0.f, 0.f, 0.f);
  float4 acc1 = make_float4(0.f, 0.f, 0.f, 0.f);

  for (int tile = waveId; tile < numTiles; tile += wavesPerPair) {
    int base = tile * 16;
    // ---- stage 16x128 tile into LDS ----
#if HAVE_ASYNC_LDS
    if (base + 16 <= N) {
      // full tile: ASYNCcnt DMA path, no VGPR round-trip
#pragma unroll
      for (int r = 0; r < 16; ++r) {
        __builtin_amdgcn_global_load_async_to_lds_b128(
            (anpm_gp128_t)(x + (size_t)(base + r) * 128 + lane * 4),
            (anpm_lp128_t)(tileLds + r * 132 + lane * 4),
            0, 0);
      }
      anpm_wait_async();
      __builtin_amdgcn_wave_barrier();
    } else {
      anpm_stage_tile_masked(x, tileLds, base, N, lane);
    }
#else
    anpm_stage_tile_masked(x, tileLds, base, N, lane);
#endif
    int nb = base + wavesPerPair * 16;
    if (nb < N) __builtin_prefetch(x + (size_t)nb * 128 + lane * 32, 0, 3);

    // ---- dots via WMMA: C[j][row] = sum_k w_j[k] * X[row][k] ----
    v8f c8 = {0.f, 0.f, 0.f, 0.f, 0.f, 0.f, 0.f, 0.f};
#pragma unroll
    for (int kk = 0; kk < 32; ++kk) {
      int c0 = kk * 4 + half;
      v2f aw = *reinterpret_cast<const v2f*>(wv + jj * 128 + c0);     // A: weights (row j)
      aw = aw * amsk;                                                 // zero rows j >= NW
      v2f bx = *reinterpret_cast<const v2f*>(tileLds + j * 132 + c0); // B: X^T (col = row)
      c8 = __builtin_amdgcn_wmma_f32_16x16x4_f32(
          false, aw, false, bx, (short)0, c8, false, false);
    }

    // ---- per-row coefficients (valid in lanes 0..15; row m = lane) ----
    float co0, co1;
    if (PASS == 2) {
      float s0 = c8[0] + c0h; s0 = s0 / fmaxf(fabsf(s0), 1e-12f);
      co0 = U0 / (1.f + __expf(-s0));
      float s1 = c8[1] + c1h; s1 = s1 / fmaxf(fabsf(s1), 1e-12f);
      co1 = U1 / (1.f + __expf(-s1));
    } else {
      float s0 = c8[0] + c0h; s0 = s0 / fmaxf(fabsf(s0), 1e-12f);
      float a10 = U0 / (1.f + __expf(-s0));
      float s2 = a10 * c8[2] + c2h; s2 = s2 / fmaxf(fabsf(s2), 1e-12f);
      float a20 = U0 / (1.f + __expf(-s2));
      co0 = a10 * a20;
      float s1 = c8[1] + c1h; s1 = s1 / fmaxf(fabsf(s1), 1e-12f);
      float a11 = U1 / (1.f + __expf(-s1));
      float s3 = a11 * c8[3] + c3h; s3 = s3 / fmaxf(fabsf(s3), 1e-12f);
      float a21 = U1 / (1.f + __expf(-s3));
      co1 = a11 * a21;
    }

    // ---- weighted row-sum: acc_h += coef_h[m] * X[m][:]  (readlane bcast) ----
#pragma unroll
    for (int m = 0; m < 16; ++m) {
      float f0 = __int_as_float(__builtin_amdgcn_readlane(__float_as_int(co0), m));
      float f1 = __int_as_float(__builtin_amdgcn_readlane(__float_as_int(co1), m));
      float4 v = *reinterpret_cast<const float4*>(tileLds + m * 132 + lane * 4);
      acc0.x = fmaf(f0, v.x, acc0.x); acc0.y = fmaf(f0, v.y, acc0.y);
      acc0.z = fmaf(f0, v.z, acc0.z); acc0.w = fmaf(f0, v.w, acc0.w);
      acc1.x = fmaf(f1, v.x, acc1.x); acc1.y = fmaf(f1, v.y, acc1.y);
      acc1.z = fmaf(f1, v.z, acc1.z); acc1.w = fmaf(f1, v.w, acc1.w);
    }
  }

  // ---- deterministic WG reduction -> per-WG partial vectors ----
  __syncthreads();
  float* red = sm;  // reuse tile LDS (4*256 <= 4*2112)
  *reinterpret_cast<float4*>(&red[warp * 256 + lane * 4]) = acc0;
  *reinterpret_cast<float4*>(&red[warp * 256 + 128 + lane * 4]) = acc1;
  __syncthreads();
  for (int h = 0; h < 2; ++h) {
    float s = 0.f;
    for (int w = 0; w < 4; ++w) s += red[w * 256 + h * 128 + tid];
    part[(size_t)pair * nwg * 256 + (size_t)wg * 256 + h * 128 + tid] = s;
  }
}

// ---------------------------------------------------------------------------
// Reduce pass-3 partials into g[pair][256] = concat(out_head0, out_head1).
// ---------------------------------------------------------------------------
__global__ __launch_bounds__(256) void anpm_reduce_g_kernel(
    const float* __restrict__ p3, int nwg, float* __restrict__ g)
{
  int pair = blockIdx.x, tid = threadIdx.x;
  float s = 0.f;
  const float* p = p3 + (size_t)pair * nwg * 256 + tid;
  for (int w = 0; w < nwg; ++w) s += p[(size_t)w * 256];
  g[pair * 256 + tid] = s;
}

// ---------------------------------------------------------------------------
// NTN + projection epilogue (per batch). Bilinear term read coalesced as
// s[f] = sum_e g2[e] * (sum_d W[f,d,e] g1[d]).
// ---------------------------------------------------------------------------
__global__ __launch_bounds__(256) void anpm_ntn_kernel(
    const float* __restrict__ g, const float* __restrict__ V_ntn,
    const float* __restrict__ W_ntn, const float* __restrict__ b_ntn,
    const float* __restrict__ p0, const float* __restrict__ p1,
    const float* __restrict__ p2, const float* __restrict__ p3v,
    float* __restrict__ out)
{
  int b = blockIdx.x, tid = threadIdx.x;
  __shared__ float g1[256], g2[256];
  __shared__ float pl[16][257];
  __shared__ float sv[16];
  g1[tid] = g[(0 * 8 + b) * 256 + tid];
  g2[tid] = g[(1 * 8 + b) * 256 + tid];
  __syncthreads();
  float myg2 = g2[tid];
  for (int f = 0; f < 16; ++f) {
    const float* Wf = W_ntn + (size_t)f * 65536;
    float a = 0.f;
    for (int d = 0; d < 256; ++d) a = fmaf(Wf[(size_t)d * 256 + tid], g1[d], a);
    pl[f][tid] = a * myg2 + V_ntn[f * 512 + tid] * g1[tid]
               + V_ntn[f * 512 + 256 + tid] * myg2;
  }
  __syncthreads();
  if (tid < 16) {
    float s = 0.f;
    for (int t2 = 0; t2 < 256; ++t2) s += pl[tid][t2];
    sv[tid] = s + b_ntn[tid];
  }
  __syncthreads();
  if (tid == 0) {
    float nrm = 0.f;
    for (int f = 0; f < 16; ++f) nrm += fabsf(sv[f]);
    nrm = fmaxf(nrm, 1e-12f);
    float y0[16];
    for (int f = 0; f < 16; ++f) { float v = sv[f] / nrm; y0[f] = v > 0.f ? v : 0.f; }
    float y1[8];
    for (int i = 0; i < 8; ++i) { float s = 0.f; for (int k = 0; k < 16; ++k) s += p0[i * 16 + k] * y0[k]; y1[i] = s; }
    float y2[4];
    for (int i = 0; i < 4; ++i) { float s = 0.f; for (int k = 0; k < 8; ++k) s += p1[i * 8 + k] * y1[k]; y2[i] = s; }
    float y3[2];
    for (int i = 0; i < 2; ++i) { float s = 0.f; for (int k = 0; k < 4; ++k) s += p2[i * 4 + k] * y2[k]; y3[i] = s; }
    out[b] = p3v[0] * y3[0] + p3v[1] * y3[1];
  }
}

// ---------------------------------------------------------------------------
extern "C" void kernel_launch(void* const* d_in, const int* in_sizes, int n_in,
                              void* d_out, int out_size, void* d_ws, size_t ws_size,
                              hipStream_t stream) {
  const float* x1     = (const float*)d_in[0];
  const float* x2     = (const float*)d_in[1];
  const float* W_att  = (const float*)d_in[2];
  const float* V_att  = (const float*)d_in[3];
  const float* Wt_att = (const float*)d_in[4];
  const float* U_att  = (const float*)d_in[5];
  const float* b_att  = (const float*)d_in[6];
  const float* V_ntn  = (const float*)d_in[7];
  const float* W_ntn  = (const float*)d_in[8];
  const float* b_ntn  = (const float*)d_in[9];
  const float* proj0  = (const float*)d_in[10];
  const float* proj1  = (const float*)d_in[11];
  const float* proj2  = (const float*)d_in[12];
  const float* proj3  = (const float*)d_in[13];
  float* out = (float*)d_out;

  int N = in_sizes[0] / (8 * 128);
  float invN = 1.0f / (float)N;

  // workspace carve (floats): meanPart | wAll | cAll | p2 | p3 | g
  int nwg = 64;
  auto need = [](int w) -> size_t {
    return (size_t)NPAIR * w * 128 + 8192 + 64 + 2 * (size_t)NPAIR * w * 256 + 4096;
  };
  while (nwg > 1 && need(nwg) * sizeof(float) > ws_size) nwg >>= 1;

  float* wsf = (float*)d_ws;
  float* meanPart = wsf; wsf += (size_t)NPAIR * nwg * 128;
  float* wAll = wsf;     wsf += 8192;
  float* cAll = wsf;     wsf += 64;
  float* p2 = wsf;       wsf += (size_t)NPAIR * nwg * 256;
  float* p3 = wsf;       wsf += (size_t)NPAIR * nwg * 256;
  float* gv = wsf;       wsf += 4096;

  dim3 gridPass(NPAIR * nwg);

  anpm_mean_kernel<<<gridPass, 256, 0, stream>>>(x1, x2, N, nwg, meanPart);
  anpm_stage_kernel<<<NPAIR, 128, 0, stream>>>(meanPart, nwg, 1, invN,
      W_att, V_att, Wt_att, b_att, wAll, cAll, 0);
  anpm_pass_kernel<2><<<gridPass, 128, 0, stream>>>(x1, x2, N, nwg, wAll, cAll, U_att, p2);
  anpm_stage_kernel<<<NPAIR, 128, 0, stream>>>(p2, nwg, 2, invN,
      W_att, V_att, Wt_att, b_att, wAll, cAll, 2);
  anpm_pass_kernel<3><<<gridPass, 128, 0, stream>>>(x1, x2, N, nwg, wAll, cAll, U_att, p3);
  anpm_reduce_g_kernel<<<NPAIR, 256, 0, stream>>>(p3, nwg, gv);
  anpm_ntn_kernel<<<8, 256, 0, stream>>>(gv, V_ntn, W_ntn, b_ntn,
      proj0, proj1, proj2, proj3, out);
}